// CosSimAttention_35158602285792
// MI455X (gfx1250) — compile-verified
//
#include <hip/hip_runtime.h>
#include <stdint.h>

// Problem sizes (fixed by the reference).
#define Bn 32
#define Sn 8192
#define Dn 256
#define TILE 16      // s-rows per tile (one WMMA 16x16x4 M-dim)
#define WAVES 4      // waves per workgroup (wave32)
#define BLOCK 128
#define LSTR 260     // LDS row stride in floats (256 + 4 pad -> <=2-way conflicts)

typedef float v2f __attribute__((ext_vector_type(2)));
typedef float v8f __attribute__((ext_vector_type(8)));
typedef float f4  __attribute__((ext_vector_type(4)));
// gcc-style int4, matching the async-builtin's pointee type exactly.
typedef int v4i_g __attribute__((vector_size(16)));
typedef __attribute__((address_space(1))) v4i_g* as1_v4i_ptr;
typedef __attribute__((address_space(3))) v4i_g* as3_v4i_ptr;

// Dynamic LDS layout (in floats)
#define K_OFF     0                       // 256 : k row
#define STAGE_OFF 320                     // WAVES*256 : per-wave attn partials
#define TILES_OFF 1344                    // WAVES * 2 * TILE * LSTR : xs tiles
#define SMEM_FLOATS (TILES_OFF + WAVES * 2 * TILE * LSTR)   // 34624 -> 138496 B

#if defined(__has_builtin)
#  if __has_builtin(__builtin_amdgcn_global_load_async_to_lds_b128)
#    define HAVE_ASYNC 1
#  endif
#endif

template <int N>
__device__ __forceinline__ void wait_asynccnt() {
#if defined(__has_builtin) && __has_builtin(__builtin_amdgcn_s_wait_asynccnt)
  __builtin_amdgcn_s_wait_asynccnt(N);
  asm volatile("" ::: "memory");
#else
  asm volatile("s_wait_asynccnt %0" :: "i"(N) : "memory");
#endif
}

#if defined(HAVE_ASYNC)
// Issue rows 0-7, cols 0-63 of a tile: 128 x b128 async copies, coalesced on the global side.
__device__ __forceinline__ void issue_tile_async(const float* __restrict__ g0,
                                                 float* lbuf, int lane) {
#pragma unroll
  for (int q = 0; q < 4; ++q) {
    int slot = q * 32 + lane;           // 128 float4 slots
    int row  = slot >> 4;               // 16 float4 per row
    int col  = (slot & 15) * 4;
    const float* g = g0 + row * Dn + col;
    float*       l = lbuf + row * LSTR + col;
    __builtin_amdgcn_global_load_async_to_lds_b128(
        (as1_v4i_ptr)g, (as3_v4i_ptr)l, 0, 0);
  }
}
#else
__device__ __forceinline__ void load_tile_regs(const float* __restrict__ g0,
                                               f4 st[4], int lane) {
#pragma unroll
  for (int q = 0; q < 4; ++q) {
    int slot = q * 32 + lane;
    int row = slot >> 4, col = (slot & 15) * 4;
    st[q] = *(const f4*)(g0 + row * Dn + col);
  }
}
__device__ __forceinline__ void store_tile_regs(float* lbuf, const f4 st[4], int lane) {
#pragma unroll
  for (int q = 0; q < 4; ++q) {
    int slot = q * 32 + lane;
    int row = slot >> 4, col = (slot & 15) * 4;
    *(f4*)(lbuf + row * LSTR + col) = st[q];
  }
}
#endif

// Fused per-tile work: dot + sumsq via streaming packed FMA, WMMA(B=ones)
// reduce+broadcast, native-math softabs, then rank-1 accumulation a[m]*xs[m,:].
__device__ __forceinline__ void compute_tile(const float* buf, const float* ks,
                                             const float* __restrict__ mask_row,
                                             float inv_kn, int lane,
                                             f4& acc0, f4& acc1) {
  const int r = lane & 15;      // row within tile
  const int h = lane >> 4;      // which half of d: [0,128) vs [128,256)
  const float* rowp = buf + r * LSTR + h * 128;
  const float* kp   = ks + h * 128;
  float d0 = 0, d1 = 0, d2 = 0, d3 = 0;
  float q0 = 0, q1 = 0, q2 = 0, q3 = 0;
#pragma unroll
  for (int j = 0; j < 32; ++j) {          // 128 d-elems per lane, b128 LDS loads
    f4 x  = *(const f4*)(rowp + 4 * j);
    f4 kv = *(const f4*)(kp + 4 * j);     // uniform per half -> LDS broadcast
    d0 = fmaf(x.x, kv.x, d0); d1 = fmaf(x.y, kv.y, d1);
    d2 = fmaf(x.z, kv.z, d2); d3 = fmaf(x.w, kv.w, d3);
    q0 = fmaf(x.x, x.x, q0);  q1 = fmaf(x.y, x.y, q1);
    q2 = fmaf(x.z, x.z, q2);  q3 = fmaf(x.w, x.w, q3);
  }
  // A-layout (16x4 f32): lane m -> (K0,K1), lane m+16 -> (K2,K3).  B = ones
  // => D[m,n] = full-row reduction, broadcast to all N (all lanes).
  v2f Ad; Ad.x = d0 + d2; Ad.y = d1 + d3;
  v2f Aq; Aq.x = q0 + q2; Aq.y = q1 + q3;
  v2f Bones; Bones.x = 1.0f; Bones.y = 1.0f;
  v8f Cz = {};
  v8f Ddot = __builtin_amdgcn_wmma_f32_16x16x4_f32(false, Ad, false, Bones,
                                                   (short)0, Cz, false, false);
  v8f Dsq  = __builtin_amdgcn_wmma_f32_16x16x4_f32(false, Aq, false, Bones,
                                                   (short)0, Cz, false, false);
  // D layout: VGPR j = row j (lanes 0-15) / row j+8 (lanes 16-31).
  // Branchless softabs(cos) for this half's 8 rows, all with native V_RSQ/V_SQRT:
  //   1/max(sqrt(ss),1e-8) == min(rsq(ss), 1e8)
  float av[8];
#pragma unroll
  for (int m = 0; m < 8; ++m) {
    float inv_xn = __builtin_fminf(__builtin_amdgcn_rsqf(Dsq[m]), 1e8f);
    float c      = Ddot[m] * inv_xn * inv_kn;
    av[m]        = __builtin_amdgcn_sqrtf(fmaf(c, c, 1e-3f));
  }
  // Cross-half exchange: lanes<16 hold rows 0-7, lanes>=16 hold rows 8-15.
  float alo[8], ahi[8];
  const bool hi = (lane >= 16);
#pragma unroll
  for (int m = 0; m < 8; ++m) {
    float other = __shfl_xor(av[m], 16);
    alo[m] = hi ? other : av[m];
    ahi[m] = hi ? av[m] : other;
  }
  // Mask for the 16 rows (uniform address -> scalar-friendly 4x float4).
  const f4* mr = (const f4*)mask_row;
  f4 mk0 = mr[0], mk1 = mr[1], mk2 = mr[2], mk3 = mr[3];
  float mk[16] = {mk0.x, mk0.y, mk0.z, mk0.w, mk1.x, mk1.y, mk1.z, mk1.w,
                  mk2.x, mk2.y, mk2.z, mk2.w, mk3.x, mk3.y, mk3.z, mk3.w};
  // Phase B: attn[d] += (a*mask + 1e-14) * xs[m,d]; lane owns d=4l..4l+3 and +128.
#pragma unroll
  for (int m = 0; m < 16; ++m) {
    float am = (m < 8) ? alo[m] : ahi[m - 8];
    float a  = fmaf(am, mk[m], 1e-14f);
    f4 xa = *(const f4*)(buf + m * LSTR + 4 * lane);
    f4 xb = *(const f4*)(buf + m * LSTR + 128 + 4 * lane);
    acc0 += a * xa;
    acc1 += a * xb;
  }
}

__global__ __launch_bounds__(BLOCK) void cosattn_pass1(
    const float* __restrict__ k, const float* __restrict__ xs,
    const float* __restrict__ mask, float* __restrict__ ws, int CH) {
  extern __shared__ float smem[];
  const int tid    = threadIdx.x;
  const int b      = blockIdx.x / CH;
  const int chunk  = blockIdx.x % CH;
  const int schunk = Sn / CH;
  const int ntiles = schunk / (WAVES * TILE);
  const int lane   = tid & 31;
  const int w      = __builtin_amdgcn_readfirstlane(tid >> 5);

  // Stage k row into LDS (cooperative, coalesced).
  float* ks = smem + K_OFF;
  {
    const float* kg = k + (size_t)b * Dn;
    ks[2 * tid]     = kg[2 * tid];
    ks[2 * tid + 1] = kg[2 * tid + 1];
  }
  __syncthreads();

  // 1/max(||k||,1e-8) per wave (redundant across waves; xor-shuffle tree).
  float inv_kn;
  {
    f4 a = *(const f4*)(ks + 8 * lane);
    f4 c = *(const f4*)(ks + 8 * lane + 4);
    float s = a.x * a.x + a.y * a.y + a.z * a.z + a.w * a.w +
              c.x * c.x + c.y * c.y + c.z * c.z + c.w * c.w;
    s += __shfl_xor(s, 1);  s += __shfl_xor(s, 2);  s += __shfl_xor(s, 4);
    s += __shfl_xor(s, 8);  s += __shfl_xor(s, 16);
    float kn = __builtin_fmaxf(__builtin_amdgcn_sqrtf(s), 1e-8f);
    inv_kn = __builtin_amdgcn_rcpf(kn);
  }

  float* buf0 = smem + TILES_OFF + (w * 2 + 0) * TILE * LSTR;
  float* buf1 = smem + TILES_OFF + (w * 2 + 1) * TILE * LSTR;

  const int s_base    = chunk * schunk + w * (schunk / WAVES);
  const float* xs_b   = xs + (size_t)b * Sn * Dn;
  const float* mask_b = mask + (size_t)b * Sn;

  f4 acc0 = {0.f, 0.f, 0.f, 0.f}, acc1 = {0.f, 0.f, 0.f, 0.f};

#if defined(HAVE_ASYNC)
  // Double-buffered async global->LDS stream, synced per-wave with ASYNCcnt.
  issue_tile_async(xs_b + (size_t)s_base * Dn, buf0, lane);
  for (int t = 0; t < ntiles; ++t) {
    float* bufc = (t & 1) ? buf1 : buf0;
    float* bufn = (t & 1) ? buf0 : buf1;
    if (t + 1 < ntiles) {
      issue_tile_async(xs_b + (size_t)(s_base + (t + 1) * TILE) * Dn, bufn, lane);
      wait_asynccnt<32>();   // each tile is 4 ops/wave, so <=32 does not force current done
    } else {
      wait_asynccnt<0>();
    }
    compute_tile(bufc, ks, mask_b + s_base + t * TILE, inv_kn, lane, acc0, acc1);
  }
#else
  // Fallback: register-staged copy (global b128 loads for t+1 overlap compute of t).
  {
    f4 stg[4];
    load_tile_regs(xs_b + (size_t)s_base * Dn, stg, lane);
    for (int t = 0; t < ntiles; ++t) {
      float* bufc = (t & 1) ? buf1 : buf0;
      store_tile_regs(bufc, stg, lane);
      if (t + 1 < ntiles)
        load_tile_regs(xs_b + (size_t)(s_base + (t + 1) * TILE) * Dn, stg, lane);
      compute_tile(bufc, ks, mask_b + s_base + t * TILE, inv_kn, lane, acc0, acc1);
    }
  }
#endif

  // Deterministic in-block reduction of the 4 per-wave attn partials.
  float* st = smem + STAGE_OFF + w * Dn;
  *(f4*)(st + 4 * lane)       = acc0;
  *(f4*)(st + 128 + 4 * lane) = acc1;
  __syncthreads();
#pragma unroll
  for (int rep = 0; rep < 2; ++rep) {
    int d = tid + rep * 128;
    float s = smem[STAGE_OFF + 0 * Dn + d] + smem[STAGE_OFF + 1 * Dn + d] +
              smem[STAGE_OFF + 2 * Dn + d] + smem[STAGE_OFF + 3 * Dn + d];
    ws[((size_t)b * CH + chunk) * Dn + d] = s;
  }
}

__global__ __launch_bounds__(256) void cosattn_pass2(const float* __restrict__ ws,
                                                     float* __restrict__ out, int CH) {
  int b = blockIdx.x;
  int d = threadIdx.x;
  float s = 0.f;
  for (int c = 0; c < CH; ++c) s += ws[((size_t)b * CH + c) * Dn + d];
  out[(size_t)b * Dn + d] = s;
}

extern "C" void kernel_launch(void* const* d_in, const int* in_sizes, int n_in,
                              void* d_out, int out_size, void* d_ws, size_t ws_size,
                              hipStream_t stream) {
  (void)in_sizes; (void)n_in; (void)out_size;
  const float* k    = (const float*)d_in[0];
  const float* xs   = (const float*)d_in[1];
  const float* mask = (const float*)d_in[2];
  float* out = (float*)d_out;

  // Pick chunk count that fits the provided workspace (power of two, >=1).
  int CH = 16;
  while (CH > 1 && ws_size < (size_t)Bn * CH * Dn * sizeof(float)) CH >>= 1;
  size_t shmem = (size_t)SMEM_FLOATS * sizeof(float);

  if (ws_size >= (size_t)Bn * CH * Dn * sizeof(float)) {
    float* ws = (float*)d_ws;
    hipLaunchKernelGGL(cosattn_pass1, dim3(Bn * CH), dim3(BLOCK), shmem, stream,
                       k, xs, mask, ws, CH);
    hipLaunchKernelGGL(cosattn_pass2, dim3(Bn), dim3(Dn), 0, stream, ws, out, CH);
  } else {
    // No usable workspace: CH=1 partial layout == output layout; write out directly.
    hipLaunchKernelGGL(cosattn_pass1, dim3(Bn), dim3(BLOCK), shmem, stream,
                       k, xs, mask, out, 1);
  }
}